// GRASSEncoder_60576218743334
// MI455X (gfx1250) — compile-verified
//
#include <hip/hip_runtime.h>
#include <hip/hip_bf16.h>
#include <math.h>

typedef __attribute__((ext_vector_type(2))) float v2f;
typedef __attribute__((ext_vector_type(8))) float v8f;

#define GRID   128
#define BLOCK  256
#define KSPLIT 4
#define FDIM   1024
#define HDIM   2048
#define BATCH  256
#define NOPS   60
#define NBOX   32
#define NSYM   16
#define RESMAX 32

__device__ __forceinline__ int iclamp(int v, int lo, int hi) {
  return v < lo ? lo : (v > hi ? hi : v);
}

__device__ __forceinline__ v8f wmma4(v2f a, v2f b, v8f c) {
  // D = A(16x4 f32, rows broadcast) * B(4x16 f32) + C, exact fp32
  return __builtin_amdgcn_wmma_f32_16x16x4_f32(false, a, false, b, (short)0, c,
                                               false, false);
}

// Matvec partial: out[n0+lanelo] += sum over K=[kbase, kbase+4*steps) of
// xs[k] * W[k*ldw + n]. xs lives in LDS. A-matrix rows are broadcast so every
// row of D equals the matvec result; lane<16 reads acc[0] = col n0+lane.
__device__ __forceinline__ float dot_wmma4(const float* __restrict__ W, int ldw,
                                           const float* __restrict__ xs,
                                           int kbase, int steps, int n0, int lane) {
  const int lanelo = lane & 15;
  const int ksel   = (lane & 16) ? 2 : 0;           // A/B layout: upper half lanes hold K+2,K+3
  const float* wp = W + (size_t)(kbase + ksel) * (size_t)ldw + (size_t)(n0 + lanelo);
  const float* xp = xs + kbase + ksel;
  const size_t wstep = (size_t)4 * (size_t)ldw;
  v8f a0 = {}, a1 = {}, a2 = {}, a3 = {};
  for (int i = 0; i < steps; i += 4) {
    v2f a, b;
    a.x = xp[0]; a.y = xp[1]; b.x = wp[0]; b.y = wp[ldw];
    a0 = wmma4(a, b, a0); xp += 4; wp += wstep;
    a.x = xp[0]; a.y = xp[1]; b.x = wp[0]; b.y = wp[ldw];
    a1 = wmma4(a, b, a1); xp += 4; wp += wstep;
    a.x = xp[0]; a.y = xp[1]; b.x = wp[0]; b.y = wp[ldw];
    a2 = wmma4(a, b, a2); xp += 4; wp += wstep;
    a.x = xp[0]; a.y = xp[1]; b.x = wp[0]; b.y = wp[ldw];
    a3 = wmma4(a, b, a3); xp += 4; wp += wstep;
  }
  v8f s = (a0 + a1) + (a2 + a3);
  return s[0];
}

__device__ __forceinline__ float dot_wmma1(const float* __restrict__ W, int ldw,
                                           const float* __restrict__ xs,
                                           int steps, int n0, int lane) {
  const int lanelo = lane & 15;
  const int ksel   = (lane & 16) ? 2 : 0;
  const float* wp = W + (size_t)ksel * (size_t)ldw + (size_t)(n0 + lanelo);
  const float* xp = xs + ksel;
  v8f a0 = {};
  for (int i = 0; i < steps; ++i) {
    v2f a, b;
    a.x = xp[0]; a.y = xp[1]; b.x = wp[0]; b.y = wp[ldw];
    a0 = wmma4(a, b, a0);
    xp += 4; wp += (size_t)4 * (size_t)ldw;
  }
  return a0[0];
}

// Monotonic-counter grid barrier (sense-free; counter never resets).
__device__ __forceinline__ void grid_sync(unsigned* cnt, unsigned* gen) {
  __threadfence();          // release this thread's global stores to device scope
  __syncthreads();
  if (threadIdx.x == 0) {
    unsigned target = *gen + GRID;
    __hip_atomic_fetch_add(cnt, 1u, __ATOMIC_ACQ_REL, __HIP_MEMORY_SCOPE_AGENT);
    while (__hip_atomic_load(cnt, __ATOMIC_ACQUIRE, __HIP_MEMORY_SCOPE_AGENT) < target)
      __builtin_amdgcn_s_sleep(2);
  }
  *gen += GRID;
  __syncthreads();
  __threadfence();          // acquire: invalidate so we see other blocks' stores
}

// Materialize a stack entry (1024 floats) into LDS. Computed results are kept
// as KSPLIT partial sums; bias + tanh is applied here at consumption time.
__device__ void stage_entry(float* dst, int tag, const float* __restrict__ boxes,
                            const float* __restrict__ respart,
                            const float* __restrict__ adj_b2,
                            const float* __restrict__ sym_b2) {
  const int kind = tag >> 16;
  const int slot = tag & 0xffff;
  if (kind == 0) {
    const float* src = boxes + (size_t)slot * FDIM;
    for (int j = threadIdx.x; j < FDIM; j += BLOCK) dst[j] = src[j];
  } else {
    const float* rp = respart + (size_t)slot * (KSPLIT * FDIM);
    const float* b2 = (kind == 1) ? adj_b2 : sym_b2;
    for (int j = threadIdx.x; j < FDIM; j += BLOCK)
      dst[j] = tanhf(rp[j] + rp[FDIM + j] + rp[2 * FDIM + j] + rp[3 * FDIM + j] + b2[j]);
  }
}

__global__ void grass_init(unsigned* cnt) { *cnt = 0u; }

__global__ void __launch_bounds__(BLOCK, 1) grass_kernel(
    const float* __restrict__ inputStacks, const float* __restrict__ symStacks,
    const int* __restrict__ ops,
    const float* __restrict__ box_W, const float* __restrict__ box_b,
    const float* __restrict__ adj_Wl, const float* __restrict__ adj_bl,
    const float* __restrict__ adj_Wr, const float* __restrict__ adj_W2,
    const float* __restrict__ adj_b2,
    const float* __restrict__ sym_Wl, const float* __restrict__ sym_bl,
    const float* __restrict__ sym_Wr, const float* __restrict__ sym_br,
    const float* __restrict__ sym_W2, const float* __restrict__ sym_b2,
    float* __restrict__ out, unsigned* cnt,
    float* __restrict__ boxes, float* __restrict__ hpart,
    float* __restrict__ respart) {
  __shared__ float lds[HDIM + 16];
  const int tid  = threadIdx.x;
  const int lane = tid & 31;
  const int wv   = tid >> 5;
  const int gw   = wv * GRID + (int)blockIdx.x;   // global wave id, 0..1023
  unsigned gen = 0;

  // ---- Phase 0: box encoder for all 32 leaves of batch 0 (32768 outputs) ----
  {
    int g  = (int)blockIdx.x * BLOCK + tid;       // GRID*BLOCK == 32768 exactly
    int nb = g >> 10, f = g & 1023;
    const float* xin = inputStacks + (size_t)nb * (BATCH * 12);
    float s = box_b[f];
#pragma unroll
    for (int d = 0; d < 12; ++d) s += xin[d] * box_W[d * FDIM + f];
    boxes[g] = tanhf(s);
  }
  grid_sync(cnt, &gen);

  // ---- Replicated scalar state machine (identical in every thread/block) ----
  int stag[24];    // stack tags: kind<<16 | slot   (kind 0=box,1=adj,2=sym)
  int ytag[4];     // sym-stack tags: row index into symmetryStacks
  stag[0] = stag[1] = 0;
  ytag[0] = ytag[1] = 0;
  int sptr = 2, yptr = 2, bptr = NBOX - 1, qptr = NSYM - 1, nres = 0;

  for (int t = 0; t < NOPS; ++t) {
    const int op = ops[t * BATCH];                // batch-0 op, uniform
    if (op <= 1) {                                // push box (and maybe sym)
      stag[iclamp(sptr, 0, 23)] = iclamp(bptr, 0, NBOX - 1);
      ++sptr; --bptr;
      if (op == 1) {
        ytag[iclamp(yptr, 0, 3)] = iclamp(qptr, 0, NSYM - 1);
        ++yptr; --qptr;
      }
      continue;
    }
    const int slot = (nres < RESMAX) ? nres : RESMAX - 1;
    ++nres;

    if (op == 2) {  // AdjEncoder: tanh(tanh(sec@Wl + bl + top@Wr)@W2 + b2)
      stage_entry(lds,        stag[iclamp(sptr - 2, 0, 23)], boxes, respart, adj_b2, sym_b2);
      stage_entry(lds + FDIM, stag[iclamp(sptr - 1, 0, 23)], boxes, respart, adj_b2, sym_b2);
      __syncthreads();
      if (gw < 128 * KSPLIT) {                    // h partials: 128 N-chunks x 4 K-splits
        int chunk = gw >> 2, split = gw & 3;
        int n0 = chunk * 16, kb = split * 256;
        float r = dot_wmma4(adj_Wl, HDIM, lds,        kb, 64, n0, lane)
                + dot_wmma4(adj_Wr, HDIM, lds + FDIM, kb, 64, n0, lane);
        if (lane < 16) hpart[split * HDIM + n0 + lane] = r;
      }
      grid_sync(cnt, &gen);
      for (int j = tid; j < HDIM; j += BLOCK)     // finalize h into LDS
        lds[j] = tanhf(hpart[j] + hpart[HDIM + j] + hpart[2 * HDIM + j] +
                       hpart[3 * HDIM + j] + adj_bl[j]);
      __syncthreads();
      if (gw < 64 * KSPLIT) {                     // out partials: 64 chunks x 4 splits
        int chunk = gw >> 2, split = gw & 3;
        int n0 = chunk * 16, kb = split * 512;
        float r = dot_wmma4(adj_W2, FDIM, lds, kb, 128, n0, lane);
        if (lane < 16)
          respart[(size_t)slot * (KSPLIT * FDIM) + split * FDIM + n0 + lane] = r;
      }
      grid_sync(cnt, &gen);
      stag[iclamp(sptr - 2, 0, 23)] = (1 << 16) | slot;
      --sptr;
    } else {        // SymEncoder: tanh(tanh(top@sWl + sbl + stop@sWr + sbr)@sW2 + sb2)
      stage_entry(lds, stag[iclamp(sptr - 1, 0, 23)], boxes, respart, adj_b2, sym_b2);
      {
        int q = ytag[iclamp(yptr - 1, 0, 3)];
        const float* sp = symStacks + (size_t)q * (BATCH * 8);
        if (tid < 8) lds[HDIM + tid] = sp[tid];
      }
      __syncthreads();
      if (gw < 128 * KSPLIT) {
        int chunk = gw >> 2, split = gw & 3;
        int n0 = chunk * 16, kb = split * 256;
        float r = dot_wmma4(sym_Wl, HDIM, lds, kb, 64, n0, lane);
        if (split == 0) r += dot_wmma1(sym_Wr, HDIM, lds + HDIM, 2, n0, lane);
        if (lane < 16) hpart[split * HDIM + n0 + lane] = r;
      }
      grid_sync(cnt, &gen);
      for (int j = tid; j < HDIM; j += BLOCK)
        lds[j] = tanhf(hpart[j] + hpart[HDIM + j] + hpart[2 * HDIM + j] +
                       hpart[3 * HDIM + j] + sym_bl[j] + sym_br[j]);
      __syncthreads();
      if (gw < 64 * KSPLIT) {
        int chunk = gw >> 2, split = gw & 3;
        int n0 = chunk * 16, kb = split * 512;
        float r = dot_wmma4(sym_W2, FDIM, lds, kb, 128, n0, lane);
        if (lane < 16)
          respart[(size_t)slot * (KSPLIT * FDIM) + split * FDIM + n0 + lane] = r;
      }
      grid_sync(cnt, &gen);
      stag[iclamp(sptr - 1, 0, 23)] = (2 << 16) | slot;
      --yptr;
    }
  }

  // ---- Final: materialize stack top -> d_out (1024 floats, blocks 0..3) ----
  if (blockIdx.x < 4) {
    int g = (int)blockIdx.x * BLOCK + tid;        // 0..1023
    int tag = stag[iclamp(sptr - 1, 0, 23)];
    int kind = tag >> 16, slot = tag & 0xffff;
    float v;
    if (kind == 0) {
      v = boxes[(size_t)slot * FDIM + g];
    } else {
      const float* rp = respart + (size_t)slot * (KSPLIT * FDIM);
      const float* b2 = (kind == 1) ? adj_b2 : sym_b2;
      v = tanhf(rp[g] + rp[FDIM + g] + rp[2 * FDIM + g] + rp[3 * FDIM + g] + b2[g]);
    }
    out[g] = v;
  }
}

extern "C" void kernel_launch(void* const* d_in, const int* in_sizes, int n_in,
                              void* d_out, int out_size, void* d_ws, size_t ws_size,
                              hipStream_t stream) {
  const float* inputStacks = (const float*)d_in[0];
  const float* symStacks   = (const float*)d_in[1];
  const int*   ops         = (const int*)  d_in[2];
  const float* box_W  = (const float*)d_in[3];
  const float* box_b  = (const float*)d_in[4];
  const float* adj_Wl = (const float*)d_in[5];
  const float* adj_bl = (const float*)d_in[6];
  const float* adj_Wr = (const float*)d_in[7];
  const float* adj_W2 = (const float*)d_in[8];
  const float* adj_b2 = (const float*)d_in[9];
  const float* sym_Wl = (const float*)d_in[10];
  const float* sym_bl = (const float*)d_in[11];
  const float* sym_Wr = (const float*)d_in[12];
  const float* sym_br = (const float*)d_in[13];
  const float* sym_W2 = (const float*)d_in[14];
  const float* sym_b2 = (const float*)d_in[15];

  unsigned* cnt  = (unsigned*)d_ws;
  float* wsf     = (float*)d_ws;
  float* boxes   = wsf + 16;                       // 32*1024
  float* hpart   = boxes + NBOX * FDIM;            // KSPLIT*2048
  float* respart = hpart + KSPLIT * HDIM;          // RESMAX*KSPLIT*1024

  grass_init<<<1, 1, 0, stream>>>(cnt);
  grass_kernel<<<GRID, BLOCK, 0, stream>>>(
      inputStacks, symStacks, ops, box_W, box_b,
      adj_Wl, adj_bl, adj_Wr, adj_W2, adj_b2,
      sym_Wl, sym_bl, sym_Wr, sym_br, sym_W2, sym_b2,
      (float*)d_out, cnt, boxes, hpart, respart);
}